// GNNRank_6356551598164
// MI455X (gfx1250) — compile-verified
//
#include <hip/hip_runtime.h>
#include <hip/hip_bf16.h>

typedef __attribute__((ext_vector_type(2))) float v2f;
typedef __attribute__((ext_vector_type(8))) float v8f;

#define D 64
#define TAU 0.5f

// ---------------------------------------------------------------------------
// SpMM: Y[row[e], :] += val[e] * X[col[e], :]   (Y pre-zeroed)
// 16 lanes per edge, each lane handles features f, f+16, f+32, f+48.
// ---------------------------------------------------------------------------
__global__ __launch_bounds__(256) void spmm_kernel(
    const int* __restrict__ row, const int* __restrict__ col,
    const float* __restrict__ val, const float* __restrict__ X,
    float* __restrict__ Y, int nedges)
{
    long long t = (long long)blockIdx.x * blockDim.x + threadIdx.x;
    int e = (int)(t >> 4);
    int f = (int)(t & 15);
    if (e >= nedges) return;
    int r = row[e];
    int c = col[e];
    float w = val[e];
    const float* xs = X + (size_t)c * D;
    float* yd = Y + (size_t)r * D;
#pragma unroll
    for (int j = 0; j < 4; ++j) {
        int idx = f + j * 16;
        atomicAdd(yd + idx, w * xs[idx]);
    }
}

// ---------------------------------------------------------------------------
// X = X + relu(AX @ W^T) using V_WMMA_F32_16X16X4_F32.
// One wave per 16-row tile of AX.
//
// A (16x4 f32) fragment layout (ISA 7.12.2): lanes 0-15 hold M=0..15 with
// VGPR0=K0,VGPR1=K1; lanes 16-31 hold K2,K3. So lane l holds
// A[row0 + l%16][k0 + (l/16)*2 + {0,1}]  -> one v2f load per fragment.
// B (4x16 f32) mirrors: lane l holds B[k0 + (l/16)*2 + {0,1}][n0 + l%16]
// = W[n0 + l%16][k0 + (l/16)*2 + {0,1}]  -> one v2f load per fragment.
// C/D (16x16 f32): vgpr i, lanes 0-15 -> M=i, N=lane; lanes 16-31 -> M=i+8.
// ---------------------------------------------------------------------------
__global__ __launch_bounds__(256) void linear_relu_res_kernel(
    const float* __restrict__ AX, const float* __restrict__ W,
    float* __restrict__ X, int nrows)
{
    const int lane = threadIdx.x & 31;
    const int wave = blockIdx.x * (blockDim.x >> 5) + (threadIdx.x >> 5);
    const int row0 = wave * 16;
    if (row0 >= nrows) return;
    const bool full_tile = (row0 + 16 <= nrows);

    const int m  = lane & 15;          // M (for A) / N (for B) within tile
    const int kh = (lane >> 4) << 1;   // 0 or 2: K sub-offset for this half-wave

    // Preload the full 16x64 A tile as 16 fragments (32 floats/lane).
    const int arow = full_tile ? (row0 + m)
                               : ((row0 + m < nrows) ? (row0 + m) : (nrows - 1));
    const float* ap = AX + (size_t)arow * D + kh;
    v2f a[16];
#pragma unroll
    for (int f = 0; f < 16; ++f)
        a[f] = *(const v2f*)(ap + 4 * f);

#pragma unroll
    for (int nt = 0; nt < 4; ++nt) {
        const int n0 = nt * 16;
        const float* wp = W + (size_t)(n0 + m) * D + kh;
        v8f c = {};
#pragma unroll
        for (int f = 0; f < 16; ++f) {
            v2f b = *(const v2f*)(wp + 4 * f);
            c = __builtin_amdgcn_wmma_f32_16x16x4_f32(
                    /*neg_a=*/false, a[f], /*neg_b=*/false, b,
                    /*c_mod=*/(short)0, c, /*reuse_a=*/false, /*reuse_b=*/false);
        }
        // Residual + ReLU writeback per C/D layout.
        const int nc    = n0 + m;
        const int rbase = row0 + ((lane >> 4) << 3);
        float* xp = X + (size_t)rbase * D + nc;
        if (full_tile) {
            // Branch-free fast path: clause the 8 loads, then add+store.
            float xr[8];
#pragma unroll
            for (int i = 0; i < 8; ++i)
                xr[i] = xp[(size_t)i * D];
#pragma unroll
            for (int i = 0; i < 8; ++i) {
                float h = c[i];
                xp[(size_t)i * D] = xr[i] + fmaxf(h, 0.0f);
            }
        } else {
#pragma unroll
            for (int i = 0; i < 8; ++i) {
                if (rbase + i < nrows) {
                    float h = c[i];
                    xp[(size_t)i * D] += fmaxf(h, 0.0f);
                }
            }
        }
    }
}

// ---------------------------------------------------------------------------
// s[i] = dot(X[i,:], Ws)
// ---------------------------------------------------------------------------
__global__ __launch_bounds__(256) void score_kernel(
    const float* __restrict__ X, const float* __restrict__ Ws,
    float* __restrict__ s, int n)
{
    int i = blockIdx.x * blockDim.x + threadIdx.x;
    if (i >= n) return;
    const float4* xr = (const float4*)(X + (size_t)i * D);
    const float4* wr = (const float4*)Ws;
    float acc = 0.f;
#pragma unroll
    for (int j = 0; j < D / 4; ++j) {
        float4 a = xr[j];
        float4 b = wr[j];
        acc += a.x * b.x + a.y * b.y + a.z * b.z + a.w * b.w;
    }
    s[i] = acc;
}

// ---------------------------------------------------------------------------
// SpMV: u[row[e]] += val[e] * v[col[e]]   (u pre-zeroed)
// ---------------------------------------------------------------------------
__global__ __launch_bounds__(256) void spmv_kernel(
    const int* __restrict__ row, const int* __restrict__ col,
    const float* __restrict__ val, const float* __restrict__ v,
    float* __restrict__ u, int nedges)
{
    int e = blockIdx.x * blockDim.x + threadIdx.x;
    if (e >= nedges) return;
    atomicAdd(u + row[e], val[e] * v[col[e]]);
}

// ---------------------------------------------------------------------------
// w[i] = u[i] - TAU*sign(u[i]); normsq += sum(w^2) (wave reduce + 1 atomic)
// ---------------------------------------------------------------------------
__global__ __launch_bounds__(256) void shrink_kernel(
    const float* __restrict__ u, float* __restrict__ w,
    float* __restrict__ normsq, int n)
{
    int i = blockIdx.x * blockDim.x + threadIdx.x;
    float x = 0.f;
    if (i < n) {
        float ui = u[i];
        float sg = (ui > 0.f) ? 1.f : ((ui < 0.f) ? -1.f : 0.f);
        x = ui - TAU * sg;
        w[i] = x;
    }
    float p = x * x;
#pragma unroll
    for (int off = 16; off > 0; off >>= 1)
        p += __shfl_down(p, off, 32);
    if ((threadIdx.x & 31) == 0)
        atomicAdd(normsq, p);
}

__global__ __launch_bounds__(256) void scale_kernel(
    float* __restrict__ w, const float* __restrict__ normsq, int n)
{
    int i = blockIdx.x * blockDim.x + threadIdx.x;
    if (i >= n) return;
    float nrm = sqrtf(*normsq);
    nrm = fmaxf(nrm, 1e-12f);
    w[i] = w[i] / nrm;
}

// ---------------------------------------------------------------------------
extern "C" void kernel_launch(void* const* d_in, const int* in_sizes, int n_in,
                              void* d_out, int out_size, void* d_ws, size_t ws_size,
                              hipStream_t stream)
{
    const int*   A_row = (const int*)  d_in[0];
    const int*   A_col = (const int*)  d_in[1];
    const float* A_val = (const float*)d_in[2];
    const int*   L_row = (const int*)  d_in[3];
    const int*   L_col = (const int*)  d_in[4];
    const float* L_val = (const float*)d_in[5];
    const float* embed = (const float*)d_in[6];
    const float* W1    = (const float*)d_in[7];
    const float* W2    = (const float*)d_in[8];
    const float* Ws    = (const float*)d_in[9];

    const int E = in_sizes[0];
    const int N = in_sizes[6] / D;

    float* Xcur   = (float*)d_ws;
    float* AX     = Xcur + (size_t)N * D;
    float* u      = AX   + (size_t)N * D;
    float* v      = u + N;
    float* normsq = v + N;

    // X = embed (never mutate inputs)
    hipMemcpyAsync(Xcur, embed, (size_t)N * D * sizeof(float),
                   hipMemcpyDeviceToDevice, stream);

    const float* Wl[2] = { W1, W2 };
    const int spmm_blocks = (int)(((long long)E * 16 + 255) / 256);
    const int waves       = (N + 15) / 16;
    const int lin_blocks  = (waves + 7) / 8;   // 8 waves per 256-thread block

    for (int l = 0; l < 2; ++l) {
        hipMemsetAsync(AX, 0, (size_t)N * D * sizeof(float), stream);
        spmm_kernel<<<spmm_blocks, 256, 0, stream>>>(A_row, A_col, A_val, Xcur, AX, E);
        linear_relu_res_kernel<<<lin_blocks, 256, 0, stream>>>(AX, Wl[l], Xcur, N);
    }

    score_kernel<<<(N + 255) / 256, 256, 0, stream>>>(Xcur, Ws, v, N);

    for (int it = 0; it < 10; ++it) {
        hipMemsetAsync(u, 0, (size_t)N * sizeof(float), stream);
        spmv_kernel<<<(E + 255) / 256, 256, 0, stream>>>(L_row, L_col, L_val, v, u, E);
        hipMemsetAsync(normsq, 0, sizeof(float), stream);
        shrink_kernel<<<(N + 255) / 256, 256, 0, stream>>>(u, v, normsq, N);
        scale_kernel<<<(N + 255) / 256, 256, 0, stream>>>(v, normsq, N);
    }

    hipMemcpyAsync(d_out, v, (size_t)N * sizeof(float),
                   hipMemcpyDeviceToDevice, stream);
}